// GATConv_17635135717523
// MI455X (gfx1250) — compile-verified
//
#include <hip/hip_runtime.h>
#include <hip/hip_bf16.h>

typedef float v2f __attribute__((ext_vector_type(2)));
typedef float v8f __attribute__((ext_vector_type(8)));

#define IN_F   256
#define HO     256
#define NHEAD  8
#define OUTF   32
#define LDSTR  260   // padded LDS row stride (floats), even, %64==4 -> conflict-free

// ---------- monotonic float<->uint encoding for atomic max ----------
__device__ inline unsigned f32_to_ordered(float f) {
    unsigned u = __float_as_uint(f);
    return (u & 0x80000000u) ? ~u : (u | 0x80000000u);
}
__device__ inline float ordered_to_f32(unsigned u) {
    return __uint_as_float((u & 0x80000000u) ? (u & 0x7FFFFFFFu) : ~u);
}

// ---------- kernel 0: init out=bias, emax=encoded(-huge), denom=0 ----------
__global__ void gat_init(float* __restrict__ out, const float* __restrict__ bias,
                         unsigned* __restrict__ emax_u, float* __restrict__ denom,
                         int N) {
    int idx = blockIdx.x * blockDim.x + threadIdx.x;
    int total = N * HO;
    for (int i = idx; i < total; i += gridDim.x * blockDim.x) {
        out[i] = bias[i & (HO - 1)];
    }
    int nh = N * NHEAD;
    for (int i = idx; i < nh; i += gridDim.x * blockDim.x) {
        emax_u[i] = 0u;       // below any encoded real float
        denom[i]  = 0.0f;
    }
}

// ---------- kernel 1: ft = feat @ W via V_WMMA_F32_16X16X4_F32 ----------
// block = 256 threads (8 waves). Block b computes rows [16b,16b+16) x all 256 cols.
// Wave w handles column tiles at w*32 and w*32+16.
__global__ __launch_bounds__(256)
void gat_gemm(const float* __restrict__ feat, const float* __restrict__ W,
              float* __restrict__ ft) {
    __shared__ float alds[16 * LDSTR];

    const int tid  = threadIdx.x;
    const int row0 = blockIdx.x * 16;

    // cooperative, coalesced load of A tile 16x256 into LDS
    for (int idx = tid; idx < 16 * IN_F; idx += 256) {
        int r = idx >> 8;
        int c = idx & (IN_F - 1);
        alds[r * LDSTR + c] = feat[(row0 + r) * IN_F + c];
    }
    __syncthreads();

    const int lane = tid & 31;
    const int wave = tid >> 5;
    const int l15  = lane & 15;
    const int half = lane >> 4;          // 0: lanes 0-15, 1: lanes 16-31
    const int col0 = wave * 32;          // two 16-wide tiles: col0, col0+16

    v8f acc0 = {};
    v8f acc1 = {};

    for (int k = 0; k < IN_F; k += 4) {
        const int kb = k + 2 * half;     // per-lane K pair {kb, kb+1}

        // A fragment 16x4 f32: lane (l15,half) holds row l15, K={kb,kb+1}
        v2f a;
        a.x = alds[l15 * LDSTR + kb];
        a.y = alds[l15 * LDSTR + kb + 1];

        // B fragment 4x16 f32: lane holds col (col0+l15), K={kb,kb+1}
        v2f b0, b1;
        b0.x = W[(kb    ) * HO + col0 + l15];
        b0.y = W[(kb + 1) * HO + col0 + l15];
        b1.x = W[(kb    ) * HO + col0 + 16 + l15];
        b1.y = W[(kb + 1) * HO + col0 + 16 + l15];

        acc0 = __builtin_amdgcn_wmma_f32_16x16x4_f32(false, a, false, b0,
                                                     (short)0, acc0, false, false);
        acc1 = __builtin_amdgcn_wmma_f32_16x16x4_f32(false, a, false, b1,
                                                     (short)0, acc1, false, false);
    }

    // C/D layout: VGPR j -> row j (lanes 0-15) / row j+8 (lanes 16-31), lane -> col
    #pragma unroll
    for (int j = 0; j < 8; ++j) {
        int row = row0 + j + 8 * half;
        ft[row * HO + col0 + l15]      = acc0[j];
        ft[row * HO + col0 + 16 + l15] = acc1[j];
    }
}

// ---------- kernel 2: el/er per (node, head) ----------
__global__ void gat_attn_logits(const float* __restrict__ ft,
                                const float* __restrict__ attn_l,
                                const float* __restrict__ attn_r,
                                float* __restrict__ el, float* __restrict__ er,
                                int N) {
    int idx = blockIdx.x * blockDim.x + threadIdx.x;
    int total = N * NHEAD;
    for (int i = idx; i < total; i += gridDim.x * blockDim.x) {
        int n = i >> 3;
        int h = i & (NHEAD - 1);
        const float* f = ft + n * HO + h * OUTF;
        const float* al = attn_l + h * OUTF;
        const float* ar = attn_r + h * OUTF;
        float sl = 0.0f, sr = 0.0f;
        #pragma unroll
        for (int o = 0; o < OUTF; ++o) {
            float v = f[o];
            sl += v * al[o];
            sr += v * ar[o];
        }
        el[i] = sl;
        er[i] = sr;
    }
}

// ---------- kernel 3: e = leaky_relu(el[src]+er[dst]); segment max ----------
__global__ void gat_edge_logits(const int* __restrict__ src, const int* __restrict__ dst,
                                const float* __restrict__ el, const float* __restrict__ er,
                                float* __restrict__ ebuf, unsigned* __restrict__ emax_u,
                                int E) {
    int idx = blockIdx.x * blockDim.x + threadIdx.x;
    int total = E * NHEAD;
    if (idx >= total) return;
    int e = idx >> 3;
    int h = idx & (NHEAD - 1);
    int s = src[e];
    int d = dst[e];
    float v = el[s * NHEAD + h] + er[d * NHEAD + h];
    v = (v >= 0.0f) ? v : 0.2f * v;
    ebuf[idx] = v;
    atomicMax(&emax_u[d * NHEAD + h], f32_to_ordered(v));
}

// ---------- kernel 4: ee = exp(e - emax[dst]); segment sum (denom) ----------
__global__ void gat_edge_exp(const int* __restrict__ dst,
                             float* __restrict__ ebuf,
                             const unsigned* __restrict__ emax_u,
                             float* __restrict__ denom, int E) {
    int idx = blockIdx.x * blockDim.x + threadIdx.x;
    int total = E * NHEAD;
    if (idx >= total) return;
    int e = idx >> 3;
    int h = idx & (NHEAD - 1);
    int d = dst[e];
    float m  = ordered_to_f32(emax_u[d * NHEAD + h]);
    float ee = __expf(ebuf[idx] - m);
    ebuf[idx] = ee;
    unsafeAtomicAdd(&denom[d * NHEAD + h], ee);
}

// ---------- kernel 5: out[dst] += (ee/denom[dst]) * ft[src]  (one block/edge) ----------
__global__ __launch_bounds__(256)
void gat_aggregate(const int* __restrict__ src, const int* __restrict__ dst,
                   const float* __restrict__ ft, const float* __restrict__ ebuf,
                   const float* __restrict__ denom, float* __restrict__ out) {
    int e = blockIdx.x;
    int c = threadIdx.x;          // 0..255
    int h = c >> 5;
    int s = src[e];
    int d = dst[e];
    float a = ebuf[e * NHEAD + h] / denom[d * NHEAD + h];
    unsafeAtomicAdd(&out[d * HO + c], ft[s * HO + c] * a);
}

extern "C" void kernel_launch(void* const* d_in, const int* in_sizes, int n_in,
                              void* d_out, int out_size, void* d_ws, size_t ws_size,
                              hipStream_t stream) {
    const float* feat   = (const float*)d_in[0];
    const float* W      = (const float*)d_in[1];
    const float* attn_l = (const float*)d_in[2];
    const float* attn_r = (const float*)d_in[3];
    const float* bias   = (const float*)d_in[4];
    const int*   src    = (const int*)d_in[5];
    const int*   dst    = (const int*)d_in[6];
    float* out = (float*)d_out;

    const int N = in_sizes[0] / IN_F;
    const int E = in_sizes[5];

    // workspace carving (256B aligned)
    char* ws = (char*)d_ws;
    size_t off = 0;
    auto carve = [&](size_t bytes) {
        void* p = ws + off;
        off += (bytes + 255) & ~(size_t)255;
        return p;
    };
    float*    ft     = (float*)   carve((size_t)N * HO * sizeof(float));
    float*    el     = (float*)   carve((size_t)N * NHEAD * sizeof(float));
    float*    er     = (float*)   carve((size_t)N * NHEAD * sizeof(float));
    unsigned* emax_u = (unsigned*)carve((size_t)N * NHEAD * sizeof(unsigned));
    float*    denom  = (float*)   carve((size_t)N * NHEAD * sizeof(float));
    float*    ebuf   = (float*)   carve((size_t)E * NHEAD * sizeof(float));
    (void)ws_size;

    gat_init<<<1024, 256, 0, stream>>>(out, bias, emax_u, denom, N);

    gat_gemm<<<N / 16, 256, 0, stream>>>(feat, W, ft);

    gat_attn_logits<<<(N * NHEAD + 255) / 256, 256, 0, stream>>>(ft, attn_l, attn_r,
                                                                 el, er, N);

    int eh_blocks = (E * NHEAD + 255) / 256;
    gat_edge_logits<<<eh_blocks, 256, 0, stream>>>(src, dst, el, er, ebuf, emax_u, E);
    gat_edge_exp<<<eh_blocks, 256, 0, stream>>>(dst, ebuf, emax_u, denom, E);

    gat_aggregate<<<E, 256, 0, stream>>>(src, dst, ft, ebuf, denom, out);
}